// LSTM_22376779612536
// MI455X (gfx1250) — compile-verified
//
#include <hip/hip_runtime.h>
#include <hip/hip_bf16.h>
#include <math.h>

typedef float v2f __attribute__((ext_vector_type(2)));
typedef float v8f __attribute__((ext_vector_type(8)));

#define TT   512
#define BB   512
#define NIN  64
#define H1   100
#define H2   50
#define H3   25
#define G1   400      // 4*H1
#define G2   200      // 4*H2
#define G3   100      // 4*H3
#define G1S  400      // gates1 row stride
#define G2S  208      // gates2 row stride (13 tiles * 16)
#define G3S  112      // gates3 row stride (7 tiles * 16)
#define SH1  104      // h1 row stride (pad 100 -> 104)
#define SH2  52       // h2 row stride (pad 50 -> 52, zeros in pad); 52=4*13 -> conflict-free
#define SH3  28       // h3 row stride (pad 25 -> 28, zeros in pad); 28=4*7  -> conflict-free
#define NT   256
#define NWAVES 8

// K-step counts (K/4, using padded K where needed)
#define KS_X1 16      // NIN/4
#define KS_H1 25      // H1/4
#define KS_H2 13      // SH2/4 (K=50 padded to 52)
#define KS_H3 7       // SH3/4 (K=25 padded to 28)

__device__ __forceinline__ float sigf(float x) { return 1.0f / (1.0f + __expf(-x)); }

__device__ __forceinline__ v8f wmma4(v2f a, v2f b, v8f c) {
  // D(16x16,f32) += A(16x4,f32) * B(4x16,f32)
  return __builtin_amdgcn_wmma_f32_16x16x4_f32(
      false, a, false, b, (short)0, c, false, false);
}

// Load KS fragments (k, k+1 at k-offset khalf within each group of 4) into registers.
template<int KS>
__device__ __forceinline__ void load_frags(v2f* f, const float* p, int khalf) {
  #pragma unroll
  for (int i = 0; i < KS; ++i) {
    f[i].x = p[4 * i + khalf];
    f[i].y = p[4 * i + khalf + 1];
  }
}

template<int KS>
__device__ __forceinline__ v8f wmma_chain(const v2f* a, const v2f* b, v8f acc) {
  #pragma unroll
  for (int i = 0; i < KS; ++i) acc = wmma4(a[i], b[i], acc);
  return acc;
}

__global__ __launch_bounds__(NT, 2) void lstm3_fused_kernel(
    const float* __restrict__ x,
    const float* __restrict__ w_ih1, const float* __restrict__ w_hh1,
    const float* __restrict__ b_ih1, const float* __restrict__ b_hh1,
    const float* __restrict__ w_ih2, const float* __restrict__ w_hh2,
    const float* __restrict__ b_ih2, const float* __restrict__ b_hh2,
    const float* __restrict__ w_ih3, const float* __restrict__ w_hh3,
    const float* __restrict__ b_ih3, const float* __restrict__ b_hh3,
    const float* __restrict__ w_lin, const float* __restrict__ b_lin,
    float* __restrict__ out)
{
  __shared__ float gates1[16 * G1S];     //  25600 B
  __shared__ float gates2[16 * G2S];     //  13312 B
  __shared__ float gates3[16 * G3S];     //   7168 B
  __shared__ float h1s[16 * SH1];        //   6656 B
  __shared__ float h2s[16 * SH2];        //   3328 B
  __shared__ float h3s[16 * SH3];        //   1792 B
  __shared__ float whh1s[G1 * H1];       // 160000 B  (400x100, exact, stride 100 conflict-free)
  __shared__ float whh2s[208 * SH2];     //  43264 B  (200x50 -> 208x52, zero padded)
  __shared__ float wih3s[112 * SH2];     //  23296 B  (100x50 -> 112x52, zero padded)
  __shared__ float whh3s[112 * SH3];     //  12544 B  (100x25 -> 112x28, zero padded)
                                         // total 296,960 B (< 320 KB WGP LDS)

  const int tid     = threadIdx.x;
  const int lane    = tid & 31;
  const int wave    = tid >> 5;
  const int col     = lane & 15;           // A-row (M) / B-col (N) index within tile
  const int khalf   = (lane >> 4) << 1;    // 0 (lanes 0-15) or 2 (lanes 16-31)
  const int rowhalf = (lane >> 4) << 3;    // C/D: rows 0-7 vs 8-15
  const int bbase   = blockIdx.x * 16;

  // ---- zero LDS (h state pads + weight pads) ----
  for (int i = tid; i < 16 * SH1; i += NT) h1s[i] = 0.f;
  for (int i = tid; i < 16 * SH2; i += NT) h2s[i] = 0.f;
  for (int i = tid; i < 16 * SH3; i += NT) h3s[i] = 0.f;
  for (int i = tid; i < 208 * SH2; i += NT) whh2s[i] = 0.f;
  for (int i = tid; i < 112 * SH2; i += NT) wih3s[i] = 0.f;
  for (int i = tid; i < 112 * SH3; i += NT) whh3s[i] = 0.f;
  __syncthreads();

  // ---- stage recurrent / layer-3 weights into LDS ----
  for (int i = tid; i < G1 * H1; i += NT)    // w_hh1 [400,100], no padding needed
    whh1s[i] = w_hh1[i];
  for (int i = tid; i < G2 * H2; i += NT) {  // w_hh2 [200,50]
    int r = i / H2, c = i - r * H2;
    whh2s[r * SH2 + c] = w_hh2[i];
  }
  for (int i = tid; i < G3 * H2; i += NT) {  // w_ih3 [100,50]
    int r = i / H2, c = i - r * H2;
    wih3s[r * SH2 + c] = w_ih3[i];
  }
  for (int i = tid; i < G3 * H3; i += NT) {  // w_hh3 [100,25]
    int r = i / H3, c = i - r * H3;
    whh3s[r * SH3 + c] = w_hh3[i];
  }

  // ---- hoist per-cell bias sums (time-invariant) into registers ----
  float bs1[7][4], bs2[4][4], bs3[2][4];
  #pragma unroll
  for (int r = 0; r < 7; ++r) {
    const int idx = tid + r * NT;
    if (idx < 16 * H1) {
      const int j = idx >> 4;
      #pragma unroll
      for (int g = 0; g < 4; ++g) bs1[r][g] = b_ih1[g * H1 + j] + b_hh1[g * H1 + j];
    }
  }
  #pragma unroll
  for (int r = 0; r < 4; ++r) {
    const int idx = tid + r * NT;
    if (idx < 16 * H2) {
      const int j = idx >> 4;
      #pragma unroll
      for (int g = 0; g < 4; ++g) bs2[r][g] = b_ih2[g * H2 + j] + b_hh2[g * H2 + j];
    }
  }
  #pragma unroll
  for (int r = 0; r < 2; ++r) {
    const int idx = tid + r * NT;
    if (idx < 16 * H3) {
      const int j = idx >> 4;
      #pragma unroll
      for (int g = 0; g < 4; ++g) bs3[r][g] = b_ih3[g * H3 + j] + b_hh3[g * H3 + j];
    }
  }

  // ---- hoist linear-head weights (uniform -> scalar regs) ----
  float wl[H3];
  #pragma unroll
  for (int j = 0; j < H3; ++j) wl[j] = w_lin[j];
  const float blin = b_lin[0];

  float c1r[7], c2r[4], c3r[2];
  #pragma unroll
  for (int r = 0; r < 7; ++r) c1r[r] = 0.f;
  #pragma unroll
  for (int r = 0; r < 4; ++r) c2r[r] = 0.f;
  #pragma unroll
  for (int r = 0; r < 2; ++r) c3r[r] = 0.f;

  __syncthreads();

  for (int t = 0; t < TT; ++t) {
    // ================= Layer 1: gates1 = x_t @ w_ih1^T + h1 @ w_hh1^T =========
    {
      // prefetch next step's x tile (16 rows x 64 floats, contiguous 4KB)
      if (t + 1 < TT && tid < 16)
        __builtin_prefetch(x + ((size_t)(t + 1) * BB + bbase + tid) * NIN, 0, 3);

      v2f ax[KS_X1], ah[KS_H1];
      load_frags<KS_X1>(ax, x + ((size_t)t * BB + bbase + col) * NIN, khalf);
      load_frags<KS_H1>(ah, &h1s[col * SH1], khalf);
      for (int tile = wave; tile < 25; tile += NWAVES) {   // wave-uniform
        const int n = tile * 16 + col;
        v2f bi[KS_X1], bh[KS_H1];
        load_frags<KS_X1>(bi, w_ih1 + (size_t)n * NIN, khalf);   // global (L2-hot), batched
        load_frags<KS_H1>(bh, &whh1s[n * H1], khalf);            // LDS, conflict-free
        v8f acc = {};
        acc = wmma_chain<KS_X1>(ax, bi, acc);
        acc = wmma_chain<KS_H1>(ah, bh, acc);
        #pragma unroll
        for (int r = 0; r < 8; ++r)
          gates1[(r + rowhalf) * G1S + n] = acc[r];
      }
    }
    __syncthreads();

    // ---------------- Layer 1 cell update ----------------
    #pragma unroll
    for (int r = 0; r < 7; ++r) {
      const int idx = tid + r * NT;
      if (idx < 16 * H1) {
        const int m = idx & 15, j = idx >> 4;
        float gi = gates1[m * G1S + j]          + bs1[r][0];
        float gf = gates1[m * G1S + H1 + j]     + bs1[r][1];
        float gg = gates1[m * G1S + 2 * H1 + j] + bs1[r][2];
        float go = gates1[m * G1S + 3 * H1 + j] + bs1[r][3];
        float c  = sigf(gf) * c1r[r] + sigf(gi) * tanhf(gg);
        c1r[r]   = c;
        h1s[m * SH1 + j] = sigf(go) * tanhf(c);
      }
    }
    __syncthreads();

    // ================= Layer 2: gates2 = h1_t @ w_ih2^T + h2 @ w_hh2^T ========
    {
      v2f a1[KS_H1], a2[KS_H2];
      load_frags<KS_H1>(a1, &h1s[col * SH1], khalf);
      load_frags<KS_H2>(a2, &h2s[col * SH2], khalf);
      for (int tile = wave; tile < 13; tile += NWAVES) {
        const int n  = tile * 16 + col;
        const int nc = (n < G2) ? n : (G2 - 1);   // clamp: discarded cols may be garbage
        v2f bi[KS_H1], bh[KS_H2];
        load_frags<KS_H1>(bi, w_ih2 + (size_t)nc * H1, khalf);   // global, K=100 exact
        load_frags<KS_H2>(bh, &whh2s[n * SH2], khalf);           // LDS, zero-padded
        v8f acc = {};
        acc = wmma_chain<KS_H1>(a1, bi, acc);
        acc = wmma_chain<KS_H2>(a2, bh, acc);
        if (n < G2) {
          #pragma unroll
          for (int r = 0; r < 8; ++r)
            gates2[(r + rowhalf) * G2S + n] = acc[r];
        }
      }
    }
    __syncthreads();

    // ---------------- Layer 2 cell update ----------------
    #pragma unroll
    for (int r = 0; r < 4; ++r) {
      const int idx = tid + r * NT;
      if (idx < 16 * H2) {
        const int m = idx & 15, j = idx >> 4;
        float gi = gates2[m * G2S + j]          + bs2[r][0];
        float gf = gates2[m * G2S + H2 + j]     + bs2[r][1];
        float gg = gates2[m * G2S + 2 * H2 + j] + bs2[r][2];
        float go = gates2[m * G2S + 3 * H2 + j] + bs2[r][3];
        float c  = sigf(gf) * c2r[r] + sigf(gi) * tanhf(gg);
        c2r[r]   = c;
        h2s[m * SH2 + j] = sigf(go) * tanhf(c);
      }
    }
    __syncthreads();

    // ================= Layer 3: gates3 = h2_t @ w_ih3^T + h3 @ w_hh3^T ========
    {
      v2f a2[KS_H2], a3[KS_H3];
      load_frags<KS_H2>(a2, &h2s[col * SH2], khalf);
      load_frags<KS_H3>(a3, &h3s[col * SH3], khalf);
      for (int tile = wave; tile < 7; tile += NWAVES) {
        const int n = tile * 16 + col;
        v2f bi[KS_H2], bh[KS_H3];
        load_frags<KS_H2>(bi, &wih3s[n * SH2], khalf);   // LDS, zero-padded rows/cols
        load_frags<KS_H3>(bh, &whh3s[n * SH3], khalf);
        v8f acc = {};
        acc = wmma_chain<KS_H2>(a2, bi, acc);
        acc = wmma_chain<KS_H3>(a3, bh, acc);
        if (n < G3) {
          #pragma unroll
          for (int r = 0; r < 8; ++r)
            gates3[(r + rowhalf) * G3S + n] = acc[r];
        }
      }
    }
    __syncthreads();

    // ---------------- Layer 3 cell update ----------------
    #pragma unroll
    for (int r = 0; r < 2; ++r) {
      const int idx = tid + r * NT;
      if (idx < 16 * H3) {
        const int m = idx & 15, j = idx >> 4;
        float gi = gates3[m * G3S + j]          + bs3[r][0];
        float gf = gates3[m * G3S + H3 + j]     + bs3[r][1];
        float gg = gates3[m * G3S + 2 * H3 + j] + bs3[r][2];
        float go = gates3[m * G3S + 3 * H3 + j] + bs3[r][3];
        float c  = sigf(gf) * c3r[r] + sigf(gi) * tanhf(gg);
        c3r[r]   = c;
        h3s[m * SH3 + j] = sigf(go) * tanhf(c);
      }
    }
    __syncthreads();

    // ================= Linear head: out[t,b] = h3 . w_lin + b_lin =============
    if (tid < 16) {
      const float* hr = &h3s[tid * SH3];
      float acc = blin;
      #pragma unroll
      for (int j = 0; j < H3; ++j) acc += hr[j] * wl[j];
      out[(size_t)t * BB + bbase + tid] = acc;
    }
    __syncthreads();
  }
}

extern "C" void kernel_launch(void* const* d_in, const int* in_sizes, int n_in,
                              void* d_out, int out_size, void* d_ws, size_t ws_size,
                              hipStream_t stream) {
  (void)in_sizes; (void)n_in; (void)d_ws; (void)ws_size; (void)out_size;
  const float* x     = (const float*)d_in[0];
  const float* w_ih1 = (const float*)d_in[1];
  const float* w_hh1 = (const float*)d_in[2];
  const float* b_ih1 = (const float*)d_in[3];
  const float* b_hh1 = (const float*)d_in[4];
  const float* w_ih2 = (const float*)d_in[5];
  const float* w_hh2 = (const float*)d_in[6];
  const float* b_ih2 = (const float*)d_in[7];
  const float* b_hh2 = (const float*)d_in[8];
  const float* w_ih3 = (const float*)d_in[9];
  const float* w_hh3 = (const float*)d_in[10];
  const float* b_ih3 = (const float*)d_in[11];
  const float* b_hh3 = (const float*)d_in[12];
  const float* w_lin = (const float*)d_in[13];
  const float* b_lin = (const float*)d_in[14];
  float* out = (float*)d_out;

  dim3 grid(BB / 16);   // 32 workgroups, one 16-batch tile each
  dim3 block(NT);       // 256 threads = 8 wave32
  lstm3_fused_kernel<<<grid, block, 0, stream>>>(
      x, w_ih1, w_hh1, b_ih1, b_hh1,
      w_ih2, w_hh2, b_ih2, b_hh2,
      w_ih3, w_hh3, b_ih3, b_hh3,
      w_lin, b_lin, out);
}